// Transformer_82961588289946
// MI455X (gfx1250) — compile-verified
//
#include <hip/hip_runtime.h>
#include <hip/hip_bf16.h>

// ---------------------------------------------------------------------------
// Tiny 1-layer attention-only transformer forward (D=32, V=64, N=4, B=262144).
// Kernel 1: build shared circuit tables (EQKE/EQKP/EVOU/PVOU/EqU) into d_ws.
// Kernel 2: stream batch; tables live in LDS; attn @ OV done with
//           v_wmma_f32_16x16x4_f32 (4 batch elems -> block-diagonal 16x16 A),
//           accumulator pre-loaded with the direct path EqU[id_n].
// HBM traffic ~ 4MB ids + 256MB out + ~25MB LDS staging => ~12us at 23.3TB/s.
// ---------------------------------------------------------------------------

typedef __attribute__((ext_vector_type(2))) float v2f;
typedef __attribute__((ext_vector_type(8))) float v8f;

#define LUT_EQKE 0      // [64][64]
#define LUT_EQKP 4096   // [64][4]
#define LUT_EVOU 4352   // [64][64]
#define LUT_PVOU 8448   // [4][64]
#define LUT_EQU  8704   // [64][64]
#define LUT_SIZE 12800  // floats (50 KB)

#define GPW 16          // groups (of 4 batch elems) per wave

__global__ __launch_bounds__(256) void precompute_kernel(
    const float* __restrict__ E, const float* __restrict__ P,
    const float* __restrict__ Pq, const float* __restrict__ Q,
    const float* __restrict__ Kw, const float* __restrict__ V,
    const float* __restrict__ O, const float* __restrict__ U,
    float* __restrict__ ws)
{
  __shared__ float sQKt[32 * 32];   // Q @ K^T
  __shared__ float sVO[32 * 32];    // V @ O
  __shared__ float sTmp[64 * 32];   // E_q @ QKt
  __shared__ float sVOU[32 * 64];   // V @ O @ U
  __shared__ float sPhat[4 * 32];   // P - P_query
  __shared__ float sPqMean[32];
  __shared__ float sPavg;
  const int t = threadIdx.x;

  if (t < 32) {
    float q = 0.f;
    for (int m = 0; m < 4; ++m) q += Pq[m * 32 + t];
    sPqMean[t] = q * 0.25f;
  }
  if (t == 0) {
    float s = 0.f;
    for (int i = 0; i < 128; ++i) s += P[i];
    sPavg = s * 0.25f;               // sum(P)/N_CTX (scalar, per reference)
  }
  if (t < 128) sPhat[t] = P[t] - Pq[t];

  for (int i = t; i < 1024; i += 256) {
    int r = i >> 5, c = i & 31;
    float a = 0.f, b = 0.f;
    for (int d = 0; d < 32; ++d) {
      a += Q[r * 32 + d] * Kw[c * 32 + d];  // (Q K^T)[r][c]
      b += V[r * 32 + d] * O[d * 32 + c];   // (V O)[r][c]
    }
    sQKt[i] = a;
    sVO[i] = b;
  }
  __syncthreads();

  for (int i = t; i < 2048; i += 256) {
    int v = i >> 5, d = i & 31;
    float a = 0.f;
    for (int c = 0; c < 32; ++c)
      a += (E[v * 32 + c] + sPqMean[c]) * sQKt[c * 32 + d];
    sTmp[i] = a;                     // (E_q QKt)[v][d]
    int r = i >> 6, u = i & 63;
    float b = 0.f;
    for (int d2 = 0; d2 < 32; ++d2)
      b += sVO[r * 32 + d2] * U[d2 * 64 + u];
    sVOU[i] = b;                     // (V O U)[r][u]
  }
  __syncthreads();

  const float pavg = sPavg;
  for (int i = t; i < 4096; i += 256) {   // EQKE = (E_q QKt) E_bar^T
    int v = i >> 6, w = i & 63;
    float a = 0.f;
    for (int d = 0; d < 32; ++d)
      a += sTmp[v * 32 + d] * (E[w * 32 + d] + pavg);
    ws[LUT_EQKE + i] = a;
  }
  for (int i = t; i < 256; i += 256) {    // EQKP = (E_q QKt) P_hat^T
    int v = i >> 2, m = i & 3;
    float a = 0.f;
    for (int d = 0; d < 32; ++d)
      a += sTmp[v * 32 + d] * sPhat[m * 32 + d];
    ws[LUT_EQKP + i] = a;
  }
  for (int i = t; i < 4096; i += 256) {   // EVOU = E_bar VOU
    int v = i >> 6, u = i & 63;
    float a = 0.f;
    for (int d = 0; d < 32; ++d)
      a += (E[v * 32 + d] + pavg) * sVOU[d * 64 + u];
    ws[LUT_EVOU + i] = a;
  }
  for (int i = t; i < 256; i += 256) {    // PVOU = P_hat VOU
    int m = i >> 6, u = i & 63;
    float a = 0.f;
    for (int d = 0; d < 32; ++d)
      a += sPhat[m * 32 + d] * sVOU[d * 64 + u];
    ws[LUT_PVOU + i] = a;
  }
  for (int i = t; i < 4096; i += 256) {   // EqU = E_q U
    int v = i >> 6, u = i & 63;
    float a = 0.f;
    for (int d = 0; d < 32; ++d)
      a += (E[v * 32 + d] + sPqMean[d]) * U[d * 64 + u];
    ws[LUT_EQU + i] = a;
  }
}

// Batched part. Each wave processes GPW groups of 4 batch elements; per group
// it builds a block-diagonal 16x16 attention matrix (rows = (b_local, n)) and
// runs 4 chained V_WMMA_F32_16X16X4_F32 per 16-wide v-tile, with the
// accumulator pre-initialized to the direct path EqU[id_n][v].
__global__ __launch_bounds__(256) void attn_kernel(
    const int* __restrict__ ids, const float* __restrict__ lut_g,
    float* __restrict__ out, int n_groups)
{
  __shared__ float lut[LUT_SIZE];
  for (int i = threadIdx.x; i < LUT_SIZE; i += 256) lut[i] = lut_g[i];
  __syncthreads();

  const int lane  = threadIdx.x & 31;
  const int wave  = blockIdx.x * (blockDim.x >> 5) + (threadIdx.x >> 5);
  const int lhalf = lane >> 4;     // 0: K'=0,1 / rows M<8 ; 1: K'=2,3 / rows M>=8
  const int lmod  = lane & 15;     // A-row M (low), B/D column N
  const int blA   = lmod >> 2;     // batch slot whose attn row this lane owns
  const int nA    = lmod & 3;      // query position n of that row

  const float* eqke = lut + LUT_EQKE;
  const float* eqkp = lut + LUT_EQKP;
  const float* evou = lut + LUT_EVOU;
  const float* pvou = lut + LUT_PVOU;
  const float* equ  = lut + LUT_EQU;
  const float scale = 0.17677669529663687f;   // 1/sqrt(32)

  for (int g = 0; g < GPW; ++g) {
    const int group = wave * GPW + g;         // wave-uniform => EXEC stays full
    if (group >= n_groups) return;
    const int b0 = group << 2;

    // Prefetch next group's ids (speculative; gfx1250 global_prefetch path).
    __builtin_prefetch(ids + (size_t)(b0 + 4) * 4, 0, 1);

    int4 idq[4];
#pragma unroll
    for (int b = 0; b < 4; ++b)
      idq[b] = reinterpret_cast<const int4*>(ids)[b0 + b];

    // ---- softmax row attn[blA][nA][0..3] (each lane owns one row) ----
    const int4 idm = (blA == 0) ? idq[0] : (blA == 1) ? idq[1]
                   : (blA == 2) ? idq[2] : idq[3];
    const int idn = (nA == 0) ? idm.x : (nA == 1) ? idm.y
                  : (nA == 2) ? idm.z : idm.w;
    float s0 = (eqke[idn * 64 + idm.x] + eqkp[idn * 4 + 0]) * scale;
    float s1 = (eqke[idn * 64 + idm.y] + eqkp[idn * 4 + 1]) * scale;
    float s2 = (eqke[idn * 64 + idm.z] + eqkp[idn * 4 + 2]) * scale;
    float s3 = (eqke[idn * 64 + idm.w] + eqkp[idn * 4 + 3]) * scale;
    float mx = fmaxf(fmaxf(s0, s1), fmaxf(s2, s3));
    float e0 = __expf(s0 - mx), e1 = __expf(s1 - mx);
    float e2 = __expf(s2 - mx), e3 = __expf(s3 - mx);
    float inv = 1.0f / (e0 + e1 + e2 + e3);
    float ar0 = e0 * inv, ar1 = e1 * inv, ar2 = e2 * inv, ar3 = e3 * inv;

    // A-matrix lane values: A(M=lmod, K'=v+2*lhalf). Zero off the diagonal blk.
    const float aL0 = lhalf ? ar2 : ar0;
    const float aL1 = lhalf ? ar3 : ar1;
    v2f A[4];
#pragma unroll
    for (int kb = 0; kb < 4; ++kb) {
      const bool on = (blA == kb);
      A[kb].x = on ? aL0 : 0.f;
      A[kb].y = on ? aL1 : 0.f;
    }

    // ---- 4 v-tiles of 16 columns each ----
#pragma unroll
    for (int vt = 0; vt < 4; ++vt) {
      const int col = vt * 16 + lmod;
      const int m0 = lhalf << 1, m1 = m0 + 1;   // B rows this lane supplies
      const float pv0 = pvou[m0 * 64 + col];
      const float pv1 = pvou[m1 * 64 + col];

      // C := direct path EqU[id_{b(M),n(M)}][col], M = r + 8*lhalf
      v8f acc;
#pragma unroll
      for (int r = 0; r < 8; ++r) {
        const int bq = r >> 2;                       // compile-time
        const int4 idb = lhalf ? idq[bq + 2] : idq[bq];
        const int nC = r & 3;                        // compile-time
        const int idc = (nC == 0) ? idb.x : (nC == 1) ? idb.y
                      : (nC == 2) ? idb.z : idb.w;
        acc[r] = equ[idc * 64 + col];
      }

      // K-dim in 4 slices: slice kb pairs batch kb's attn block with its OV.
#pragma unroll
      for (int kb = 0; kb < 4; ++kb) {
        const int4 idb = idq[kb];
        const int i0 = lhalf ? idb.z : idb.x;        // token at position m0
        const int i1 = lhalf ? idb.w : idb.y;        // token at position m1
        v2f B;
        B.x = evou[i0 * 64 + col] + pv0;             // OV[kb][m0][col]
        B.y = evou[i1 * 64 + col] + pv1;             // OV[kb][m1][col]
        acc = __builtin_amdgcn_wmma_f32_16x16x4_f32(
            false, A[kb], false, B, (short)0, acc, false, false);
      }

      // D(M, N=lmod) -> out[b0 + M/4][M%4][col]; flat = b0*256 + M*64 + col
      float* op = out + (size_t)b0 * 256 + col;
#pragma unroll
      for (int r = 0; r < 8; ++r)
        op[(r + 8 * lhalf) * 64] = acc[r];
    }
  }
}

extern "C" void kernel_launch(void* const* d_in, const int* in_sizes, int n_in,
                              void* d_out, int out_size, void* d_ws, size_t ws_size,
                              hipStream_t stream) {
  const int*   ids = (const int*)d_in[0];
  const float* E   = (const float*)d_in[1];
  const float* P   = (const float*)d_in[2];
  const float* Pq  = (const float*)d_in[3];
  const float* Q   = (const float*)d_in[4];
  const float* K   = (const float*)d_in[5];
  const float* V   = (const float*)d_in[6];
  const float* O   = (const float*)d_in[7];
  const float* U   = (const float*)d_in[8];
  float* out = (float*)d_out;
  float* ws  = (float*)d_ws;

  precompute_kernel<<<1, 256, 0, stream>>>(E, P, Pq, Q, K, V, O, U, ws);

  const int batch    = in_sizes[0] / 4;   // [B, N_CTX=4]
  const int n_groups = batch >> 2;        // 4 batch elems per WMMA group
  const int waves    = (n_groups + GPW - 1) / GPW;
  const int blocks   = (waves + 7) / 8;   // 8 waves (256 threads) per block
  attn_kernel<<<blocks, 256, 0, stream>>>(ids, ws, out, n_groups);
}